// PixelAdaptiveRefinement_300647710772
// MI455X (gfx1250) — compile-verified
//
#include <hip/hip_runtime.h>
#include <math.h>

// ---------------------------------------------------------------------------
// Static problem shapes (from reference):
//   images [4,3,512,512] f32, masks [4,21,128,128] f32, out = masks shape.
//   K = 32 neighbors (8 offsets x dilations {1,2,4,8}), 10 iterations.
// ---------------------------------------------------------------------------
#define BATCH   4
#define CIMG    3
#define CMSK    21
#define HH      128
#define WW      128
#define INH     512
#define INW     512
#define KNB     32
#define NITER   10

typedef __attribute__((ext_vector_type(2))) float v2f;
typedef __attribute__((ext_vector_type(8))) float v8f;

// ---------------------------------------------------------------------------
// gfx1250 async-to-LDS path (probe via __has_builtin; sync fallback otherwise)
// Probe result round 2: builtin exists; arg0 is non-const AS1 int* ("__device__
// int*"), so cast to the exact parameter types.
// ---------------------------------------------------------------------------
#if defined(__HIP_DEVICE_COMPILE__) && __has_builtin(__builtin_amdgcn_global_load_async_to_lds_b32)
#define HAVE_ASYNC_LDS 1
#else
#define HAVE_ASYNC_LDS 0
#endif

#if defined(__HIP_DEVICE_COMPILE__) && !HAVE_ASYNC_LDS
#warning "gfx1250 global_load_async_to_lds builtins unavailable; using sync LDS staging"
#endif

#if HAVE_ASYNC_LDS
#  if __has_builtin(__builtin_amdgcn_s_wait_asynccnt)
#    define WAIT_ASYNC(n) __builtin_amdgcn_s_wait_asynccnt(n)
#  else
#    define WAIT_ASYNC(n) asm volatile("s_wait_asynccnt %0" :: "i"(n))
#  endif
#else
#  define WAIT_ASYNC(n) do {} while (0)
#endif

typedef __attribute__((address_space(1))) int g_int;
typedef __attribute__((address_space(3))) int l_int;

__device__ __forceinline__ void async_copy_f32(const float* g, float* l) {
#if HAVE_ASYNC_LDS
    __builtin_amdgcn_global_load_async_to_lds_b32(
        (g_int*)const_cast<float*>(g),
        (l_int*)l, 0, 0);
#else
    *l = *g;
#endif
}

__device__ __forceinline__ int clampi(int v, int lo, int hi) {
    return v < lo ? lo : (v > hi ? hi : v);
}

// Neighbor offset tables (OFFS excludes center (1,1)); OI/OJ in {0,1,2}.
__device__ const int OI_TB[8] = {0,0,0,1,1,2,2,2};
__device__ const int OJ_TB[8] = {0,1,2,0,2,0,1,2};
__device__ const int DIL_TB[4] = {1,2,4,8};

// ---------------------------------------------------------------------------
// Init: build bilinear interp matrix M [128 x 512] (align_corners=True) and
// posw[32] = 0.01 * softmax(pos_aff) (compile-time-constant 32-vector).
// ---------------------------------------------------------------------------
__global__ __launch_bounds__(256) void init_kernel(float* __restrict__ M,
                                                   float* __restrict__ posw) {
    int idx = blockIdx.x * blockDim.x + threadIdx.x;   // 0 .. 128*512-1
    if (idx < HH * INH) {
        int o = idx / INH;
        int i = idx % INH;
        double coord = (double)o * (double)(INH - 1) / (double)(HH - 1);
        int i0 = (int)floor(coord);
        i0 = clampi(i0, 0, INH - 1);
        int i1 = clampi(i0 + 1, 0, INH - 1);
        float w = (float)(coord - (double)i0);
        float v = 0.0f;
        if (i == i0) v += 1.0f - w;
        if (i == i1) v += w;
        M[o * INW + i] = v;
    }
    if (blockIdx.x == 0 && threadIdx.x == 0) {
        const float S2 = 1.41421356237309515f;
        float base[8] = {S2, 1.f, S2, 1.f, 1.f, S2, 1.f, S2};
        float pv[KNB];
        for (int di = 0; di < 4; ++di)
            for (int o = 0; o < 8; ++o)
                pv[di * 8 + o] = base[o] * (float)DIL_TB[di];
        float s = 0.f, ss = 0.f;
        for (int k = 0; k < KNB; ++k) { s += pv[k]; ss += pv[k] * pv[k]; }
        float mean = s / (float)KNB;
        float var = (ss - (float)KNB * mean * mean) / (float)(KNB - 1);
        float sd = sqrtf(var > 0.f ? var : 0.f);
        float inv = 1.0f / ((sd + 1e-8f) * 0.3f);
        float pa[KNB], mx = -1e30f;
        for (int k = 0; k < KNB; ++k) {
            float t = pv[k] * inv;
            pa[k] = -(t * t);
            mx = pa[k] > mx ? pa[k] : mx;
        }
        float es = 0.f;
        for (int k = 0; k < KNB; ++k) { pa[k] = expf(pa[k] - mx); es += pa[k]; }
        for (int k = 0; k < KNB; ++k) posw[k] = 0.01f * pa[k] / es;
    }
}

// ---------------------------------------------------------------------------
// WMMA GEMM 1: tmp[m][o][j] = sum_i M[o][i] * img[m][i][j]
//   A = M (128x512), B = img map (512x512), C = tmp map (128x512), f32 WMMA.
// One wave per 16x16 output tile; K-loop of 512/4 = 128 v_wmma_f32_16x16x4_f32.
// VGPR layouts per CDNA5 ISA 7.12.2 (16x4 A / 4x16 B / 16x16 C).
// ---------------------------------------------------------------------------
__global__ __launch_bounds__(128) void gemm_resize_h(const float* __restrict__ M,
                                                     const float* __restrict__ img,
                                                     float* __restrict__ tmp) {
    const int wave = threadIdx.x >> 5;
    const int lane = threadIdx.x & 31;
    const int tile = blockIdx.x * 4 + wave;            // 12 maps * 8 * 32 tiles
    const int m  = tile / (8 * 32);
    const int t  = tile % (8 * 32);
    const int row0 = (t / 32) * 16;
    const int col0 = (t % 32) * 16;

    const int r = lane & 15;
    const int h = lane >> 4;                           // K-half select
    const float* __restrict__ B = img + (size_t)m * INH * INW;

    v8f c = {};
    for (int kk = 0; kk < INH; kk += 4) {
        const int ka = kk + h * 2;
        v2f a, b;
        a.x = M[(row0 + r) * INW + ka];
        a.y = M[(row0 + r) * INW + ka + 1];
        b.x = B[(size_t)(ka + 0) * INW + col0 + r];
        b.y = B[(size_t)(ka + 1) * INW + col0 + r];
        c = __builtin_amdgcn_wmma_f32_16x16x4_f32(false, a, false, b,
                                                  (short)0, c, false, false);
    }
    float* __restrict__ C = tmp + (size_t)m * HH * INW;
    #pragma unroll
    for (int rr = 0; rr < 8; ++rr)
        C[(row0 + rr + 8 * h) * INW + col0 + r] = c[rr];
}

// ---------------------------------------------------------------------------
// WMMA GEMM 2: small[m][o][p] = sum_j tmp[m][o][j] * M[p][j]   (B = M^T)
// ---------------------------------------------------------------------------
__global__ __launch_bounds__(128) void gemm_resize_w(const float* __restrict__ tmp,
                                                     const float* __restrict__ M,
                                                     float* __restrict__ simg) {
    const int wave = threadIdx.x >> 5;
    const int lane = threadIdx.x & 31;
    const int tile = blockIdx.x * 4 + wave;            // 12 maps * 8 * 8 tiles
    const int m  = tile / (8 * 8);
    const int t  = tile % (8 * 8);
    const int row0 = (t / 8) * 16;
    const int col0 = (t % 8) * 16;

    const int r = lane & 15;
    const int h = lane >> 4;
    const float* __restrict__ A = tmp + (size_t)m * HH * INW;

    v8f c = {};
    for (int kk = 0; kk < INW; kk += 4) {
        const int ka = kk + h * 2;
        v2f a, b;
        a.x = A[(row0 + r) * INW + ka];
        a.y = A[(row0 + r) * INW + ka + 1];
        // B[j][p] = M[p][j]  -> contiguous along j for a fixed column p
        b.x = M[(col0 + r) * INW + ka];
        b.y = M[(col0 + r) * INW + ka + 1];
        c = __builtin_amdgcn_wmma_f32_16x16x4_f32(false, a, false, b,
                                                  (short)0, c, false, false);
    }
    float* __restrict__ C = simg + (size_t)m * HH * WW;
    #pragma unroll
    for (int rr = 0; rr < 8; ++rr)
        C[(row0 + rr + 8 * h) * WW + col0 + r] = c[rr];
}

// ---------------------------------------------------------------------------
// Affinity: one thread per (b, h, w). 3x32 neighbor block held in registers;
// per-channel unbiased std over K; in-register softmax over K; add posw.
// aff layout: [b][k][h*w].
// ---------------------------------------------------------------------------
__global__ __launch_bounds__(256) void affinity_kernel(const float* __restrict__ simg,
                                                       const float* __restrict__ posw,
                                                       float* __restrict__ aff) {
    const int idx = blockIdx.x * blockDim.x + threadIdx.x;  // 0 .. 4*16384-1
    const int b = idx >> 14;
    const int p = idx & 16383;
    const int hh = p >> 7;
    const int ww = p & 127;
    const float* __restrict__ base = simg + (size_t)b * CIMG * HH * WW;

    float ctr[CIMG];
    #pragma unroll
    for (int c = 0; c < CIMG; ++c) ctr[c] = base[c * HH * WW + p];

    float nbr[CIMG][KNB];
    #pragma unroll
    for (int di = 0; di < 4; ++di) {
        const int d = DIL_TB[di];
        #pragma unroll
        for (int o = 0; o < 8; ++o) {
            const int rr = clampi(hh + (OI_TB[o] - 1) * d, 0, HH - 1);
            const int cc = clampi(ww + (OJ_TB[o] - 1) * d, 0, WW - 1);
            const int k = di * 8 + o;
            #pragma unroll
            for (int c = 0; c < CIMG; ++c)
                nbr[c][k] = base[c * HH * WW + rr * WW + cc];
        }
    }

    float inv[CIMG];
    #pragma unroll
    for (int c = 0; c < CIMG; ++c) {
        float s = 0.f, ss = 0.f;
        #pragma unroll
        for (int k = 0; k < KNB; ++k) { s += nbr[c][k]; ss += nbr[c][k] * nbr[c][k]; }
        float mean = s / (float)KNB;
        float var = (ss - (float)KNB * mean * mean) / (float)(KNB - 1);
        float sd = sqrtf(var > 0.f ? var : 0.f);
        inv[c] = 1.0f / ((sd + 1e-8f) * 0.3f);
    }

    float av[KNB];
    float mx = -1e30f;
    #pragma unroll
    for (int k = 0; k < KNB; ++k) {
        float s = 0.f;
        #pragma unroll
        for (int c = 0; c < CIMG; ++c) {
            float t = fabsf(nbr[c][k] - ctr[c]) * inv[c];
            s += t * t;
        }
        av[k] = -s * (1.0f / (float)CIMG);
        mx = av[k] > mx ? av[k] : mx;
    }
    float es = 0.f;
    #pragma unroll
    for (int k = 0; k < KNB; ++k) { av[k] = __expf(av[k] - mx); es += av[k]; }
    const float rs = 1.0f / es;

    float* __restrict__ ab = aff + (size_t)b * KNB * HH * WW + p;
    #pragma unroll
    for (int k = 0; k < KNB; ++k)
        ab[k * HH * WW] = av[k] * rs + posw[k];
}

// ---------------------------------------------------------------------------
// Propagation: out[b,c,p] = sum_k aff[b,k,p] * in[b,c,nbr_k(p)].
// Block = one 16x16 pixel tile of one batch, all 21 channels.
//  - 32 aff weights cached in registers, reused across channels (21x reuse).
//  - Per channel, the 32x32 halo tile (edge clamp applied at staging) is
//    streamed into LDS with global_load_async_to_lds_b32, double-buffered
//    across the channel loop (ASYNCcnt pipelining), then the 32 gathers per
//    pixel come from LDS (bank-conflict-free: wave = 2 rows x 16 columns).
//  - L2 gather traffic drops ~8x vs direct global gathers.
// ---------------------------------------------------------------------------
__global__ __launch_bounds__(256) void prop_kernel(const float* __restrict__ min_,
                                                   const float* __restrict__ aff,
                                                   float* __restrict__ mout) {
    __shared__ float buf[2][32 * 32];

    const int blk  = blockIdx.x;           // b*64 + tile
    const int b    = blk >> 6;
    const int tile = blk & 63;
    const int h0 = (tile >> 3) * 16;
    const int w0 = (tile & 7)  * 16;
    const int ty = threadIdx.x >> 4;
    const int tx = threadIdx.x & 15;
    const int hh = h0 + ty;
    const int ww = w0 + tx;
    const int p  = hh * WW + ww;

    // Per-pixel affinity weights, register-resident across all channels.
    float a[KNB];
    const float* __restrict__ ab = aff + (size_t)b * KNB * HH * WW + p;
    #pragma unroll
    for (int k = 0; k < KNB; ++k) a[k] = ab[k * HH * WW];

    // LDS-local gather offsets (tile already edge-clamped, so no clamp here).
    int npl[KNB];
    #pragma unroll
    for (int di = 0; di < 4; ++di) {
        const int d = DIL_TB[di];
        #pragma unroll
        for (int o = 0; o < 8; ++o) {
            const int lr = ty + (OI_TB[o] - 1) * d + 8;   // 0..31
            const int lc = tx + (OJ_TB[o] - 1) * d + 8;   // 0..31
            npl[di * 8 + o] = lr * 32 + lc;
        }
    }

    // Staging map: 4 elements per thread; element e -> clamped global offset.
    int goff[4];
    int loff[4];
    #pragma unroll
    for (int q = 0; q < 4; ++q) {
        const int e  = (int)threadIdx.x + 256 * q;        // 0..1023
        const int gr = clampi(h0 - 8 + (e >> 5), 0, HH - 1);
        const int gc = clampi(w0 - 8 + (e & 31), 0, WW - 1);
        goff[q] = gr * WW + gc;
        loff[q] = e;
    }

    const float* __restrict__ mb = min_ + (size_t)b * CMSK * HH * WW;
    float* __restrict__ ob       = mout + (size_t)b * CMSK * HH * WW;

    // Prologue: stream channel 0 into buf[0].
    {
        const float* mc = mb;
        #pragma unroll
        for (int q = 0; q < 4; ++q) async_copy_f32(mc + goff[q], &buf[0][loff[q]]);
    }

    for (int c = 0; c < CMSK; ++c) {
        if (c + 1 < CMSK) {
            // Stream next channel into the other buffer while computing this one.
            const float* mc = mb + (c + 1) * (HH * WW);
            float* lb = buf[(c + 1) & 1];
            #pragma unroll
            for (int q = 0; q < 4; ++q) async_copy_f32(mc + goff[q], &lb[loff[q]]);
            WAIT_ASYNC(4);     // async loads complete in order -> tile c is in LDS
        } else {
            WAIT_ASYNC(0);
        }
        __syncthreads();       // publish this wave's LDS writes to all 8 waves

        const float* __restrict__ lb = buf[c & 1];
        float s = 0.f;
        #pragma unroll
        for (int k = 0; k < KNB; ++k) s = fmaf(a[k], lb[npl[k]], s);
        ob[c * (HH * WW) + p] = s;

        __syncthreads();       // all reads of buf[c&1] done before it is re-targeted
    }
}

// ---------------------------------------------------------------------------
// Workspace layout (float offsets):
//   M     [0,        65536)   128x512 interp matrix
//   posw  [65536,    65568)   32 floats
//   tmp   [66560,   852992)   12x128x512 intermediate
//   simg  [860160, 1056768)   12x128x128 resized images
//   aff   [1060864,3158016)   4x32x128x128 affinities
//   bufA  [3162112,4538368)   mask ping buffer (pong buffer = d_out)
// Total ~17.3 MB.
// ---------------------------------------------------------------------------
extern "C" void kernel_launch(void* const* d_in, const int* in_sizes, int n_in,
                              void* d_out, int out_size, void* d_ws, size_t ws_size,
                              hipStream_t stream) {
    const float* images = (const float*)d_in[0];
    const float* masks  = (const float*)d_in[1];
    float* out = (float*)d_out;
    float* ws  = (float*)d_ws;

    float* M    = ws;
    float* posw = ws + 65536;
    float* tmp  = ws + 66560;
    float* simg = ws + 860160;
    float* aff  = ws + 1060864;
    float* bufA = ws + 3162112;

    // 1) interp matrix + constant pos softmax
    init_kernel<<<(HH * INH + 255) / 256, 256, 0, stream>>>(M, posw);

    // 2) resize = two f32-WMMA GEMMs
    gemm_resize_h<<<768, 128, 0, stream>>>(M, images, tmp);
    gemm_resize_w<<<192, 128, 0, stream>>>(tmp, M, simg);

    // 3) per-pixel affinities
    affinity_kernel<<<(BATCH * HH * WW + 255) / 256, 256, 0, stream>>>(simg, posw, aff);

    // 4) 10 propagation iterations, ping-pong bufA <-> d_out (iter 9 -> d_out)
    const float* cur = masks;
    for (int it = 0; it < NITER; ++it) {
        float* dst = (it & 1) ? out : bufA;
        prop_kernel<<<BATCH * 64, 256, 0, stream>>>(cur, aff, dst);
        cur = dst;
    }
}